// SparseGraphLearn_28690381537605
// MI455X (gfx1250) — compile-verified
//
#include <hip/hip_runtime.h>

typedef __attribute__((ext_vector_type(2))) float v2f;
typedef __attribute__((ext_vector_type(8))) float v8f;

#define N_NODES 8192
#define IN_DIM  256
#define OUT_DIM 128
#define N_EDGES 262144

// ---------------------------------------------------------------------------
// Kernel 1: h = X @ W  via fp32 WMMA (V_WMMA_F32_16X16X4_F32).
// One wave computes one 16x16 tile of h; K-loop of 64 wmma steps.
// 4 waves/block share the same 16 A-rows (same mt, 4 consecutive nt tiles).
// A layout (16x4 f32): lanes 0-15 M=0..15, VGPR0=K0,VGPR1=K1; lanes16-31 K2,K3.
// B layout (4x16 f32): lanes 0-15 N=0..15, VGPR0=K0,VGPR1=K1; lanes16-31 K2,K3.
// C layout: VGPR i -> M=i (lanes0-15) / M=i+8 (lanes16-31), N=lane%16.
// ---------------------------------------------------------------------------
__global__ __launch_bounds__(128) void gemm_h_wmma(const float* __restrict__ X,
                                                   const float* __restrict__ W,
                                                   float* __restrict__ H) {
  const int lane = threadIdx.x & 31;
  const int wave = threadIdx.x >> 5;
  const int tile = blockIdx.x * 4 + wave;   // 4096 tiles total
  const int mt   = tile >> 3;               // 0..511
  const int nt   = tile & 7;                // 0..7
  const int half = lane >> 4;
  const int l15  = lane & 15;
  const int arow = mt * 16 + l15;           // A: M = lane%16
  const int bcol = nt * 16 + l15;           // B: N = lane%16

  v8f acc = {};
  for (int k0 = 0; k0 < IN_DIM; k0 += 4) {
    const int k = k0 + 2 * half;
    v2f av;
    av.x = X[arow * IN_DIM + k + 0];
    av.y = X[arow * IN_DIM + k + 1];
    v2f bv;
    bv.x = W[(k + 0) * OUT_DIM + bcol];
    bv.y = W[(k + 1) * OUT_DIM + bcol];
    acc = __builtin_amdgcn_wmma_f32_16x16x4_f32(
        /*neg_a=*/false, av, /*neg_b=*/false, bv,
        /*c_mod=*/(short)0, acc, /*reuse_a=*/false, /*reuse_b=*/false);
  }

  const int mbase = mt * 16 + half * 8;
#pragma unroll
  for (int i = 0; i < 8; ++i)
    H[(mbase + i) * OUT_DIM + bcol] = acc[i];
}

// ---------------------------------------------------------------------------
// Counting sort of edges by source row.
// ---------------------------------------------------------------------------
__global__ void zero_counts(int* __restrict__ cnt) {
  int i = blockIdx.x * blockDim.x + threadIdx.x;
  if (i < N_NODES) cnt[i] = 0;
}

__global__ void count_edges(const long long* __restrict__ edge,
                            int* __restrict__ cnt) {
  int i = blockIdx.x * blockDim.x + threadIdx.x;
  if (i < N_EDGES) atomicAdd(&cnt[(int)edge[i]], 1);
}

// Exclusive prefix sum of 8192 counts, single 1024-thread block.
__global__ __launch_bounds__(1024) void scan_rows(const int* __restrict__ cnt,
                                                  int* __restrict__ off,
                                                  int* __restrict__ cur) {
  __shared__ int buf[1024];
  const int t = threadIdx.x;
  int c[8];
  int s = 0;
#pragma unroll
  for (int i = 0; i < 8; ++i) { c[i] = cnt[t * 8 + i]; s += c[i]; }
  buf[t] = s;
  __syncthreads();
  for (int o = 1; o < 1024; o <<= 1) {
    int add = (t >= o) ? buf[t - o] : 0;
    __syncthreads();
    buf[t] += add;
    __syncthreads();
  }
  int run = buf[t] - s;  // exclusive prefix for this thread's chunk
#pragma unroll
  for (int i = 0; i < 8; ++i) {
    off[t * 8 + i] = run;
    cur[t * 8 + i] = run;
    run += c[i];
  }
}

__global__ void scatter_edges(const long long* __restrict__ edge,
                              int* __restrict__ cur,
                              int* __restrict__ dst) {
  int i = blockIdx.x * blockDim.x + threadIdx.x;
  if (i < N_EDGES) {
    int r = (int)edge[i];
    int c = (int)edge[N_EDGES + i];
    int p = atomicAdd(&cur[r], 1);
    dst[p] = c;
  }
}

// ---------------------------------------------------------------------------
// Kernel: one workgroup per row. Build the dense row in LDS (ds_add_f32
// handles duplicate-edge summation), compute full-row softmax, stream it out
// with non-temporal stores.
// ---------------------------------------------------------------------------
__global__ __launch_bounds__(256) void row_softmax(
    const float* __restrict__ H, const float* __restrict__ avec,
    const int* __restrict__ row_cnt, const int* __restrict__ row_off,
    const int* __restrict__ edge_dst, float* __restrict__ SG) {
  __shared__ float rowv[N_NODES];
  __shared__ float red[256];

  const int t    = threadIdx.x;
  const int r    = blockIdx.x;
  const int lane = t & 31;
  const int wave = t >> 5;

  for (int j = t; j < N_NODES; j += 256) rowv[j] = 0.0f;

  // Per-wave preload of h[r] (128 dims over 32 lanes x 4) and a.
  const float hr0 = H[r * OUT_DIM + lane];
  const float hr1 = H[r * OUT_DIM + lane + 32];
  const float hr2 = H[r * OUT_DIM + lane + 64];
  const float hr3 = H[r * OUT_DIM + lane + 96];
  const float a0 = avec[lane];
  const float a1 = avec[lane + 32];
  const float a2 = avec[lane + 64];
  const float a3 = avec[lane + 96];
  __syncthreads();

  const int cnt   = row_cnt[r];
  const int start = row_off[r];
  for (int i = wave; i < cnt; i += 8) {
    const int c = edge_dst[start + i];
    const float* hc = H + c * OUT_DIM;  // L2-resident gather
    float p = fabsf(hr0 - hc[lane])      * a0
            + fabsf(hr1 - hc[lane + 32]) * a1
            + fabsf(hr2 - hc[lane + 64]) * a2
            + fabsf(hr3 - hc[lane + 96]) * a3;
#pragma unroll
    for (int o = 16; o > 0; o >>= 1) p += __shfl_xor(p, o, 32);
    if (lane == 0) {
      float v = p > 0.0f ? p : 0.0f;   // relu
      atomicAdd(&rowv[c], v);          // ds_add_f32: sums duplicates
    }
  }
  __syncthreads();

  // Row max (entries are >= 0, implicit zeros included).
  float m = 0.0f;
  for (int j = t; j < N_NODES; j += 256) m = fmaxf(m, rowv[j]);
  red[t] = m;
  __syncthreads();
  for (int o = 128; o > 0; o >>= 1) {
    if (t < o) red[t] = fmaxf(red[t], red[t + o]);
    __syncthreads();
  }
  m = red[0];
  __syncthreads();

  // Row sum of exp(x - m).
  float s = 0.0f;
  for (int j = t; j < N_NODES; j += 256) s += expf(rowv[j] - m);
  red[t] = s;
  __syncthreads();
  for (int o = 128; o > 0; o >>= 1) {
    if (t < o) red[t] += red[t + o];
    __syncthreads();
  }
  s = red[0];
  const float inv = 1.0f / s;

  // Streaming non-temporal write of the softmaxed row.
  float* out = SG + (size_t)r * N_NODES;
  for (int j = t; j < N_NODES; j += 256)
    __builtin_nontemporal_store(expf(rowv[j] - m) * inv, &out[j]);
}

// sgraph += dense binary adjacency (duplicates summed).
__global__ void add_adj(const long long* __restrict__ edge,
                        float* __restrict__ SG) {
  int i = blockIdx.x * blockDim.x + threadIdx.x;
  if (i < N_EDGES) {
    size_t r = (size_t)edge[i];
    size_t c = (size_t)edge[N_EDGES + i];
    atomicAdd(&SG[r * N_NODES + c], 1.0f);
  }
}

// ---------------------------------------------------------------------------
extern "C" void kernel_launch(void* const* d_in, const int* in_sizes, int n_in,
                              void* d_out, int out_size, void* d_ws, size_t ws_size,
                              hipStream_t stream) {
  const float* X       = (const float*)d_in[0];       // [8192,256]
  const float* W       = (const float*)d_in[1];       // [256,128]
  const float* avec    = (const float*)d_in[2];       // [128,1]
  const long long* edge = (const long long*)d_in[3];  // [2,E] int64

  float* H  = (float*)d_out;                          // [8192,128]
  float* SG = H + (size_t)N_NODES * OUT_DIM;          // [8192,8192]

  char* ws      = (char*)d_ws;
  int* row_cnt  = (int*)(ws);
  int* row_off  = (int*)(ws + 1 * N_NODES * sizeof(int));
  int* row_cur  = (int*)(ws + 2 * N_NODES * sizeof(int));
  int* edge_dst = (int*)(ws + 3 * N_NODES * sizeof(int));  // E ints

  zero_counts<<<N_NODES / 256, 256, 0, stream>>>(row_cnt);
  gemm_h_wmma<<<1024, 128, 0, stream>>>(X, W, H);
  count_edges<<<N_EDGES / 256, 256, 0, stream>>>(edge, row_cnt);
  scan_rows<<<1, 1024, 0, stream>>>(row_cnt, row_off, row_cur);
  scatter_edges<<<N_EDGES / 256, 256, 0, stream>>>(edge, row_cur, edge_dst);
  row_softmax<<<N_NODES, 256, 0, stream>>>(H, avec, row_cnt, row_off, edge_dst, SG);
  add_adj<<<N_EDGES / 256, 256, 0, stream>>>(edge, SG);
}